// MT2VEncoderFusion_46763603919530
// MI455X (gfx1250) — compile-verified
//
#include <hip/hip_runtime.h>

// MI455X / gfx1250, wave32. One workgroup (256 thr = 8 waves) per (b,d) series.
// GASF/GADF/outer computed as rank-1/2 outer products on the matrix pipe via
// chained V_WMMA_F32_16X16X4_F32 (K=0 slice only -> layout-robust); RP on VALU.
// Store-bandwidth-bound: 32 MB output, each element computed once.

typedef __attribute__((ext_vector_type(2))) float v2f;
typedef __attribute__((ext_vector_type(8))) float v8f;

#define B_ 32
#define L_ 128
#define D_ 16
#define M_ 4
#define EPS_ 0.001f

__global__ __launch_bounds__(256) void ts2img_kernel(const float* __restrict__ x,
                                                     const float* __restrict__ wts,
                                                     float* __restrict__ out) {
  __shared__ float s_xn[L_];   // normalized+clipped series (cos(phi))
  __shared__ float s_sn[L_];   // sqrt(1 - xn^2)            (sin(phi))
  __shared__ float s_rmin[8];
  __shared__ float s_rmax[8];

  const int bd   = blockIdx.x;        // b*D + d
  const int b    = bd >> 4;           // / D_
  const int d    = bd & (D_ - 1);
  const int tid  = threadIdx.x;
  const int lane = tid & 31;
  const int wave = tid >> 5;

  // ---- load series (stride D), min/max reduce over L=128 ----
  const int l = tid & (L_ - 1);       // waves 4-7 duplicate 0-3: harmless for min/max
  const float v = x[(b * L_ + l) * D_ + d];
  float vmin = v, vmax = v;
#pragma unroll
  for (int off = 16; off > 0; off >>= 1) {
    vmin = fminf(vmin, __shfl_xor(vmin, off, 32));
    vmax = fmaxf(vmax, __shfl_xor(vmax, off, 32));
  }
  if (lane == 0) { s_rmin[wave] = vmin; s_rmax[wave] = vmax; }
  __syncthreads();
  float gmin = s_rmin[0], gmax = s_rmax[0];
#pragma unroll
  for (int i = 1; i < 8; ++i) {
    gmin = fminf(gmin, s_rmin[i]);
    gmax = fmaxf(gmax, s_rmax[i]);
  }
  if (tid < L_) {
    float xn = 2.0f * (v - gmin) / (gmax - gmin + 1e-8f) - 1.0f;
    xn = fminf(fmaxf(xn, -1.0f + EPS_), 1.0f - EPS_);
    s_xn[tid] = xn;
    s_sn[tid] = sqrtf(1.0f - xn * xn);
  }
  __syncthreads();

  // ---- uniform method selection: argmax over M=4 weights (first-max wins) ----
  const float* wp = wts + (size_t)(b * D_ + d) * M_;
  const float w0 = wp[0], w1 = wp[1], w2 = wp[2], w3 = wp[3];
  int   idx = 0;
  float wb  = w0;
  if (w1 > wb) { wb = w1; idx = 1; }
  if (w2 > wb) { wb = w2; idx = 2; }
  if (w3 > wb) { wb = w3; idx = 3; }
  const float scale = (1.0f - wb) + wb;   // match reference fp rounding of 1-sg(w)+w

  float* __restrict__ obase = out + (size_t)(b * D_ + d) * (L_ * L_);

  if (idx == 2) {
    // Recurrence plot: 1 - |xn_i - xn_j| (not a matmul) -> VALU, b128 stores.
    float4* __restrict__ obase4 = (float4*)obase;
#pragma unroll 4
    for (int e4 = tid; e4 < (L_ * L_) / 4; e4 += 256) {
      const int e  = e4 << 2;
      const int i  = e >> 7;
      const int j  = e & (L_ - 1);
      const float xi = s_xn[i];
      float4 r;
      r.x = (1.0f - fabsf(xi - s_xn[j + 0])) * scale;
      r.y = (1.0f - fabsf(xi - s_xn[j + 1])) * scale;
      r.z = (1.0f - fabsf(xi - s_xn[j + 2])) * scale;
      r.w = (1.0f - fabsf(xi - s_xn[j + 3])) * scale;
      obase4[e4] = r;
    }
  } else {
    // Matrix-pipe path. Wave w owns tile-row w (rows w*16 .. w*16+15).
    // A fragment (16x4 f32, 2 VGPRs): lanes 0-15 hold M=lane, VGPR0=K0.
    // Only the K=0 slice is used (layout-robust); the rank-2 term is a second
    // chained WMMA with the minus sign folded into B (f32 WMMA has no A/B NEG).
    // Loads are unconditional (whole wave) + cndmask select, so EXEC stays all-1s.
    const int  ti  = wave;
    const int  sub = lane & 15;
    const bool lo  = lane < 16;

    float axv = s_xn[ti * 16 + sub];
    float asv = s_sn[ti * 16 + sub];
    axv = lo ? axv : 0.0f;               // lanes 16-31 hold K=2/3 -> zero
    asv = lo ? asv : 0.0f;

    // idx 0 GASF : D = xn_i*xn_j - s_i*s_j -> A1=xn, B1=xn ; A2=s,  B2=-s
    // idx 1 GADF : D = s_i*xn_j - xn_i*s_j -> A1=s,  B1=xn ; A2=xn, B2=-s
    // idx 3 outer: D = xn_i*xn_j           -> A1=xn, B1=xn ; A2=0 (2nd term inert)
    v2f a1 = {0.0f, 0.0f}, a2 = {0.0f, 0.0f};
    if (idx == 1)      { a1.x = asv; a2.x = axv; }
    else if (idx == 0) { a1.x = axv; a2.x = asv; }
    else               { a1.x = axv; a2.x = 0.0f; }

#pragma unroll
    for (int tj = 0; tj < 8; ++tj) {
      float bxv = s_xn[tj * 16 + sub];
      float bsv = s_sn[tj * 16 + sub];
      v2f b1 = {0.0f, 0.0f}, b2 = {0.0f, 0.0f};
      b1.x = lo ?  bxv : 0.0f;
      b2.x = lo ? -bsv : 0.0f;

      v8f c = {};
      // 8 args: (neg_a, A, neg_b, B, c_mod, C, reuse_a, reuse_b)
      c = __builtin_amdgcn_wmma_f32_16x16x4_f32(false, a1, false, b1,
                                                (short)0, c, false, false);
      c = __builtin_amdgcn_wmma_f32_16x16x4_f32(false, a2, false, b2,
                                                (short)0, c, false, false);

      // C/D layout: VGPR vr -> row M = vr (+8 for lanes 16-31), N = lane%16.
      const int jcol  = tj * 16 + sub;
      const int rbase = ti * 16 + (lo ? 0 : 8);
#pragma unroll
      for (int vr = 0; vr < 8; ++vr) {
        obase[(rbase + vr) * L_ + jcol] = c[vr] * scale;
      }
    }
  }
}

extern "C" void kernel_launch(void* const* d_in, const int* in_sizes, int n_in,
                              void* d_out, int out_size, void* d_ws, size_t ws_size,
                              hipStream_t stream) {
  (void)in_sizes; (void)n_in; (void)d_ws; (void)ws_size; (void)out_size;
  const float* x   = (const float*)d_in[0];
  const float* wts = (const float*)d_in[1];
  float* out       = (float*)d_out;
  ts2img_kernel<<<dim3(B_ * D_), dim3(256), 0, stream>>>(x, wts, out);
}